// VectorQuantizer_40931038330994
// MI455X (gfx1250) — compile-verified
//
#include <hip/hip_runtime.h>

typedef float v2f __attribute__((ext_vector_type(2)));
typedef float v8f __attribute__((ext_vector_type(8)));

#define NUM_CB   2048
#define DIM      256
#define HWSZ     1024          // 32*32
#define NROWS    16384         // 16 * 1024
#define Q_ELEMS  4194304       // 16*256*32*32
#define MAIN_BLOCKS 256        // 16384 rows / 64 rows per block

// ---------------------------------------------------------------------------
// Kernel 1: c2[j] = sum_k codebook[j][k]^2   (one wave per codebook row)
// ---------------------------------------------------------------------------
__global__ void vq_c2_kernel(const float* __restrict__ cb, float* __restrict__ c2) {
    int w    = threadIdx.x >> 5;
    int lane = threadIdx.x & 31;
    int row  = blockIdx.x * 8 + w;            // 256 blocks * 8 waves = 2048 rows
    const float* p = cb + row * DIM;
    float s = 0.0f;
#pragma unroll
    for (int t = 0; t < DIM / 32; ++t) {
        float v = p[lane + 32 * t];
        s += v * v;
    }
#pragma unroll
    for (int m = 16; m >= 1; m >>= 1) s += __shfl_xor(s, m, 32);
    if (lane == 0) c2[row] = s;
}

// ---------------------------------------------------------------------------
// Kernel 2: main VQ kernel.
// Grid: 256 blocks x 128 threads (4 waves). Each wave owns 16 rows of z_flat
// and scans all 2048 codebook entries with V_WMMA_F32_16X16X4_F32.
// ---------------------------------------------------------------------------
__global__ void __launch_bounds__(128, 1)
vq_main_kernel(const float* __restrict__ z,
               const float* __restrict__ cb,
               const float* __restrict__ c2,
               float* __restrict__ qout,
               float* __restrict__ partial) {
    __shared__ int   sidx[4 * 16];
    __shared__ float ssum[4];

    const int lane = threadIdx.x & 31;
    const int w    = threadIdx.x >> 5;
    const int m0   = blockIdx.x * 64 + w * 16;   // first z_flat row of this wave
    const int bb   = m0 >> 10;                   // batch index (m0 multiple of 16)
    const int hw0  = m0 & 1023;                  // hw offset within image
    const int mrow = lane & 15;                  // row within tile / N within tile
    const int half = lane >> 4;                  // K-pair selector

    const float* zbase = z + (size_t)bb * (DIM * HWSZ) + hw0;

    // ---- preload A fragments for full K=256 (64 k-steps, v2f each) ----
    v2f a[64];
#pragma unroll
    for (int ks = 0; ks < 64; ++ks) {
        int k = 4 * ks + 2 * half;
        float x0 = zbase[(k + 0) * HWSZ + mrow];
        float x1 = zbase[(k + 1) * HWSZ + mrow];
        v2f t; t.x = x0; t.y = x1;
        a[ks] = t;
    }

    float minv[8];
    int   mini[8];
#pragma unroll
    for (int r = 0; r < 8; ++r) { minv[r] = 3.402823466e+38f; mini[r] = 0; }

    // ---- sweep codebook: 32 groups of 4 N-tiles (64 entries per group) ----
    for (int g = 0; g < 32; ++g) {
        const int nbase = g * 64;
        const float* b0 = cb + (size_t)(nbase + 0  + mrow) * DIM + 2 * half;
        const float* b1 = cb + (size_t)(nbase + 16 + mrow) * DIM + 2 * half;
        const float* b2 = cb + (size_t)(nbase + 32 + mrow) * DIM + 2 * half;
        const float* b3 = cb + (size_t)(nbase + 48 + mrow) * DIM + 2 * half;

        v8f acc0 = {}, acc1 = {}, acc2 = {}, acc3 = {};
#pragma unroll
        for (int ks = 0; ks < 64; ++ks) {
            v2f f0 = *(const v2f*)(b0 + 4 * ks);
            v2f f1 = *(const v2f*)(b1 + 4 * ks);
            v2f f2 = *(const v2f*)(b2 + 4 * ks);
            v2f f3 = *(const v2f*)(b3 + 4 * ks);
            acc0 = __builtin_amdgcn_wmma_f32_16x16x4_f32(false, a[ks], false, f0,
                                                         (short)0, acc0, false, false);
            acc1 = __builtin_amdgcn_wmma_f32_16x16x4_f32(false, a[ks], false, f1,
                                                         (short)0, acc1, false, false);
            acc2 = __builtin_amdgcn_wmma_f32_16x16x4_f32(false, a[ks], false, f2,
                                                         (short)0, acc2, false, false);
            acc3 = __builtin_amdgcn_wmma_f32_16x16x4_f32(false, a[ks], false, f3,
                                                         (short)0, acc3, false, false);
        }

        // dist = ||c||^2 - 2*(z.c); running argmin (indices increase -> strict <
        // keeps first occurrence, matching jnp.argmin)
        float c20 = c2[nbase + 0  + mrow];
        float c21 = c2[nbase + 16 + mrow];
        float c22 = c2[nbase + 32 + mrow];
        float c23 = c2[nbase + 48 + mrow];
        int n0 = nbase + 0  + mrow;
        int n1 = nbase + 16 + mrow;
        int n2 = nbase + 32 + mrow;
        int n3 = nbase + 48 + mrow;
#pragma unroll
        for (int r = 0; r < 8; ++r) {
            float d0 = c20 - 2.0f * acc0[r];
            float d1 = c21 - 2.0f * acc1[r];
            float d2 = c22 - 2.0f * acc2[r];
            float d3 = c23 - 2.0f * acc3[r];
            if (d0 < minv[r]) { minv[r] = d0; mini[r] = n0; }
            if (d1 < minv[r]) { minv[r] = d1; mini[r] = n1; }
            if (d2 < minv[r]) { minv[r] = d2; mini[r] = n2; }
            if (d3 < minv[r]) { minv[r] = d3; mini[r] = n3; }
        }
    }

    // ---- cross-lane argmin within each 16-lane group (rows M = 8*half + r) ----
#pragma unroll
    for (int r = 0; r < 8; ++r) {
#pragma unroll
        for (int m = 1; m <= 8; m <<= 1) {
            float ov = __shfl_xor(minv[r], m, 32);
            int   oi = __shfl_xor(mini[r], m, 32);
            if (ov < minv[r] || (ov == minv[r] && oi < mini[r])) {
                minv[r] = ov; mini[r] = oi;
            }
        }
    }

    // publish the 16 per-row indices of this wave via LDS
    if (lane == 0 || lane == 16) {
#pragma unroll
        for (int r = 0; r < 8; ++r) sidx[w * 16 + half * 8 + r] = mini[r];
    }
    __syncthreads();
    const int myidx = sidx[w * 16 + mrow];

    // ---- gather + straight-through write + loss accumulation ----
    const float* crow  = cb + (size_t)myidx * DIM;
    float* qbase = qout + (size_t)bb * (DIM * HWSZ) + hw0;
    float lsum = 0.0f;
#pragma unroll 4
    for (int d2 = 0; d2 < DIM; d2 += 2) {
        int d = d2 + half;                  // lanes 0-15: even d; 16-31: odd d
        float cv = crow[d];
        float zv = zbase[d * HWSZ + mrow];
        float df = cv - zv;
        lsum += df * df;
        qbase[d * HWSZ + mrow] = zv + (cv - zv);  // match reference rounding
    }

#pragma unroll
    for (int m = 16; m >= 1; m >>= 1) lsum += __shfl_xor(lsum, m, 32);
    if (lane == 0) ssum[w] = lsum;
    __syncthreads();
    if (threadIdx.x == 0)
        partial[blockIdx.x] = (ssum[0] + ssum[1]) + (ssum[2] + ssum[3]);
}

// ---------------------------------------------------------------------------
// Kernel 3: deterministic final reduction of block partials -> 3 scalars
// ---------------------------------------------------------------------------
__global__ void vq_final_kernel(const float* __restrict__ partial,
                                float* __restrict__ out) {
    float s = 0.0f;
    for (int i = 0; i < MAIN_BLOCKS; ++i) s += partial[i];
    float mse = s / (float)Q_ELEMS;
    out[Q_ELEMS + 0] = mse + 0.25f * mse;   // loss
    out[Q_ELEMS + 1] = mse;                 // codebook_loss
    out[Q_ELEMS + 2] = mse;                 // commitment_loss
}

// ---------------------------------------------------------------------------
extern "C" void kernel_launch(void* const* d_in, const int* in_sizes, int n_in,
                              void* d_out, int out_size, void* d_ws, size_t ws_size,
                              hipStream_t stream) {
    const float* z  = (const float*)d_in[0];   // (16, 256, 32, 32)
    const float* cb = (const float*)d_in[1];   // (2048, 256)
    float* out = (float*)d_out;                // q (4194304) + 3 scalars

    float* c2      = (float*)d_ws;             // 2048 floats
    float* partial = c2 + NUM_CB;              // 256 floats

    vq_c2_kernel<<<NUM_CB / 8, 256, 0, stream>>>(cb, c2);
    vq_main_kernel<<<MAIN_BLOCKS, 128, 0, stream>>>(z, cb, c2, out, partial);
    vq_final_kernel<<<1, 1, 0, stream>>>(partial, out);
}